// NeighborhoodLoss_80590766342218
// MI455X (gfx1250) — compile-verified
//
#include <hip/hip_runtime.h>
#include <hip/hip_bf16.h>

// Problem constants (match reference setup_inputs exactly).
constexpr int BATCH = 64;
constexpr int TLEN  = 8192;
constexpr int DIM   = 64;
constexpr int SUBL  = 4096;               // SUB_TRAJECTORY_LENGTH
constexpr float INHERENT_NEIGHBORHOOD = 0.5f;

constexpr int JPT   = 16;                 // timesteps (diffs) per thread
constexpr int JGRP  = SUBL / JPT;         // 256 j-groups per batch
constexpr int D4    = DIM / 4;            // 16 float4 per row
constexpr int BLOCK = 256;                // 8 waves (wave32)
constexpr int NBLK  = (BATCH * JGRP * D4) / BLOCK;  // 1024 blocks

typedef __attribute__((ext_vector_type(2))) float v2f;
typedef __attribute__((ext_vector_type(8))) float v8f;

// Wave-level sum of 32 fp32 lane partials using V_WMMA_F32_16X16X4_F32.
// Logical view: D = ones(16x4) x B(4x16). B's 64 slots hold the 32 lane
// partials (VGPR0) and zeros (VGPR1); every row of D is the column-sum
// vector, so the 16 distinct values in d[0] sum to the wave total.
// Must be called with EXEC all-ones (no divergence) per ISA 7.12.
__device__ __forceinline__ float wave_sum_wmma(float acc) {
    v2f a; a[0] = 1.0f; a[1] = 1.0f;      // A = ones(16x4)
    v2f b; b[0] = acc;  b[1] = 0.0f;      // B carries lane partials
    v8f c = {};
    v8f d = __builtin_amdgcn_wmma_f32_16x16x4_f32(
        /*neg_a=*/false, a, /*neg_b=*/false, b,
        /*c_mod=*/(short)0, c, /*reuse_a=*/false, /*reuse_b=*/false);
    float s = d[0];                       // one column-sum per lane (dup'd x2)
    s += __shfl_xor(s, 1);
    s += __shfl_xor(s, 2);
    s += __shfl_xor(s, 4);
    s += __shfl_xor(s, 8);                // sum of 16 column-sums = wave total
    return s;
}

__device__ __forceinline__ float block_sum(float acc) {
    __shared__ float lws[BLOCK / 32];
    const int wave = threadIdx.x >> 5;
    const int lane = threadIdx.x & 31;
    float w = wave_sum_wmma(acc);         // all waves, full EXEC
    if (lane == 0) lws[wave] = w;
    __syncthreads();
    float t = 0.0f;
    if (threadIdx.x == 0) {
        #pragma unroll
        for (int i = 0; i < BLOCK / 32; ++i) t += lws[i];
    }
    return t;                             // valid on thread 0 only
}

// Kernel 1: streaming squared-difference accumulation -> per-block partials.
__global__ void __launch_bounds__(BLOCK)
nbhd_loss_partials(const float* __restrict__ pred,
                   const int* __restrict__ s_ptr,
                   float* __restrict__ partial) {
    const int s = s_ptr[0];                                // device scalar
    const int t = blockIdx.x * BLOCK + threadIdx.x;
    const int d4 = t & (D4 - 1);                           // fastest: coalesce
    const int jg = (t >> 4) & (JGRP - 1);
    const int b  = t >> 12;                                // t / (JGRP*D4)

    const float4* row = (const float4*)pred
        + ((size_t)b * TLEN + (size_t)(s + jg * JPT)) * D4 + d4;

    float4 prev = row[0];
    float acc = 0.0f;
    #pragma unroll
    for (int i = 1; i <= JPT; ++i) {
        float4 cur = row[(size_t)i * D4];                  // global_load_b128
        float dx = cur.x - prev.x;
        float dy = cur.y - prev.y;
        float dz = cur.z - prev.z;
        float dw = cur.w - prev.w;
        acc = fmaf(dx, dx, acc);
        acc = fmaf(dy, dy, acc);
        acc = fmaf(dz, dz, acc);
        acc = fmaf(dw, dw, acc);
        prev = cur;                                        // row reuse: 17 loads / 16 diffs
    }

    float bs = block_sum(acc);
    if (threadIdx.x == 0) partial[blockIdx.x] = bs;
}

// Kernel 2: combine 1024 block partials -> scalar loss.
__global__ void __launch_bounds__(BLOCK)
nbhd_loss_final(const float* __restrict__ partial, float* __restrict__ out) {
    float acc = 0.0f;
    #pragma unroll
    for (int i = 0; i < NBLK / BLOCK; ++i)
        acc += partial[threadIdx.x + i * BLOCK];
    float total = block_sum(acc);
    if (threadIdx.x == 0)
        out[0] = total * (1.0f / (float)BATCH) - INHERENT_NEIGHBORHOOD;
}

extern "C" void kernel_launch(void* const* d_in, const int* in_sizes, int n_in,
                              void* d_out, int out_size, void* d_ws, size_t ws_size,
                              hipStream_t stream) {
    (void)in_sizes; (void)n_in; (void)out_size; (void)ws_size;
    const float* pred  = (const float*)d_in[0];
    const int*   s_ptr = (const int*)d_in[1];
    float*       outp  = (float*)d_out;
    float*       part  = (float*)d_ws;     // NBLK floats = 4 KB, rewritten every call

    nbhd_loss_partials<<<NBLK, BLOCK, 0, stream>>>(pred, s_ptr, part);
    nbhd_loss_final<<<1, BLOCK, 0, stream>>>(part, outp);
}